// KLinear_81312320848055
// MI455X (gfx1250) — compile-verified
//
#include <hip/hip_runtime.h>

typedef __attribute__((ext_vector_type(2))) float v2f;
typedef __attribute__((ext_vector_type(8))) float v8f;
typedef __attribute__((ext_vector_type(4))) unsigned int u32x4;
typedef __attribute__((ext_vector_type(8))) int i32x8;
typedef __attribute__((ext_vector_type(4))) int i32x4;

#define N_DIM 4096          // OUTF == INF == BATCH == 4096
#define BM 128              // block tile M (w rows)
#define BN 256              // block tile N (batch cols)
#define BK 16               // K step per LDS tile
#define LSTR 20             // LDS row stride in floats = BK + 4 pad (TDM pad reproduces this)
#define NIT (N_DIM / BK)    // 256 K iterations

#if defined(__has_builtin)
#if __has_builtin(__builtin_amdgcn_tensor_load_to_lds)
#define HAVE_TDM 1
#endif
#endif
#ifndef HAVE_TDM
#define HAVE_TDM 0
#endif

// --------------------------------------------------------------------------
// Kernel 1: materialize w = sum_k kron(a_k, b_k) * (1 / (2*sqrt(8)))
// w[i][j] = sum_k a_k[i>>sh_k][j>>sh_k] * b_k[i&m_k][j&m_k],  sh_k = log2(t_k)
// --------------------------------------------------------------------------
__global__ __launch_bounds__(256) void build_w_kernel(
    const float* __restrict__ a0, const float* __restrict__ a1,
    const float* __restrict__ a2, const float* __restrict__ a3,
    const float* __restrict__ a4, const float* __restrict__ a5,
    const float* __restrict__ a6, const float* __restrict__ a7,
    const float* __restrict__ b0, const float* __restrict__ b1,
    const float* __restrict__ b2, const float* __restrict__ b3,
    const float* __restrict__ b4, const float* __restrict__ b5,
    const float* __restrict__ b6, const float* __restrict__ b7,
    float* __restrict__ w)
{
    const float* A[8] = {a0, a1, a2, a3, a4, a5, a6, a7};
    const float* B[8] = {b0, b1, b2, b3, b4, b5, b6, b7};
    const int SH[8] = {6, 6, 7, 5, 8, 4, 9, 3};   // log2 of b_k dims

    const int idx = blockIdx.x * blockDim.x + threadIdx.x;
    const int i   = idx >> 10;
    const int j0  = (idx & 1023) << 2;

    float acc0 = 0.f, acc1 = 0.f, acc2 = 0.f, acc3 = 0.f;
#pragma unroll
    for (int k = 0; k < 8; ++k) {
        const int sh   = SH[k];
        const int t    = 1 << sh;
        const int s    = N_DIM >> sh;
        const int mask = t - 1;
        const float av = A[k][(i >> sh) * s + (j0 >> sh)];
        const float* bp = B[k] + (i & mask) * t + (j0 & mask);
        acc0 = fmaf(av, bp[0], acc0);
        acc1 = fmaf(av, bp[1], acc1);
        acc2 = fmaf(av, bp[2], acc2);
        acc3 = fmaf(av, bp[3], acc3);
    }
    const float scale = 0.17677669529663687f;  // 1 / (2*sqrt(8))
    float4 out = make_float4(acc0 * scale, acc1 * scale, acc2 * scale, acc3 * scale);
    *(float4*)(w + (size_t)i * N_DIM + j0) = out;
}

// --------------------------------------------------------------------------
// TDM helpers: issue a 2D tile load (tile_rows x BK f32, row stride N_DIM)
// into LDS with hardware padding of 4 DWORDs every 16 DWORDs (-> LSTR = 20).
// D# per CDNA5 ISA.
// --------------------------------------------------------------------------
#if HAVE_TDM
__device__ __forceinline__ void tdm_tile_load(const float* gptr, unsigned lds_byte,
                                              int tile_rows)
{
    unsigned long long ga = (unsigned long long)(size_t)gptr;
    u32x4 g0;
    g0[0] = 1u;                                       // count=1, user descriptor
    g0[1] = lds_byte;                                 // lds_addr
    g0[2] = (unsigned)(ga & 0xffffffffu);             // global_addr[31:0]
    g0[3] = (unsigned)((ga >> 32) & 0x1ffffffu)       // global_addr[56:32]
          | (2u << 30);                               // type = 2 ("image")
    i32x8 g1;
    g1[0] = (2 << 16)      // data_size = 4 bytes
          | (1 << 20)      // pad_enable
          | (3 << 22)      // pad_interval: 16 DWORDs
          | (3 << 25);     // pad_amount:   4 DWORDs
    g1[1] = (N_DIM & 0xffff) << 16;                   // tensor_dim0[15:0]
    g1[2] = (N_DIM & 0xffff) << 16;                   // dim0 hi16(=0) | tensor_dim1[15:0]
    g1[3] = (BK << 16);                               // dim1 hi16(=0) | tile_dim0
    g1[4] = tile_rows;                                // tile_dim1 (tile_dim2 = 0)
    g1[5] = N_DIM;                                    // tensor_dim0_stride[31:0]
    g1[6] = 0;
    g1[7] = 0;
    i32x4 z4 = (i32x4){0, 0, 0, 0};
#if defined(__clang_major__) && __clang_major__ >= 23
    i32x8 z8 = (i32x8){0, 0, 0, 0, 0, 0, 0, 0};
    __builtin_amdgcn_tensor_load_to_lds(g0, g1, z4, z4, z8, 0);
#else
    __builtin_amdgcn_tensor_load_to_lds(g0, g1, z4, z4, 0);
#endif
}
#endif

// --------------------------------------------------------------------------
// Kernel 2: y[i][c] = sum_j w[i][j] * x[c][j] + bias[c]
// 128x256 block tile, 8 waves, 64x64 per wave (4x4 fragments of 16x16),
// v_wmma_f32_16x16x4_f32, TDM double-buffered LDS tiles.
// --------------------------------------------------------------------------
__global__ __launch_bounds__(256) void kron_gemm_wmma(
    const float* __restrict__ W, const float* __restrict__ X,
    const float* __restrict__ bias, float* __restrict__ Y)
{
    // double buffer: [buf][ A(128 rows) then B(256 rows) ], rows of LSTR floats
    __shared__ float lds[2 * (BM + BN) * LSTR];       // 60 KB
    const int BUF = (BM + BN) * LSTR;

    const int tid  = threadIdx.x;
    const int lane = tid & 31;
    const int wv   = tid >> 5;           // 0..7
    const int wm   = (wv & 1) * 64;      // wave M offset in block tile
    const int wn   = (wv >> 1) * 64;     // wave N offset in block tile
    const int r    = lane & 15;
    const int hi   = lane >> 4;          // K pair select / C-half select

    const int bm = blockIdx.y * BM;
    const int bn = blockIdx.x * BN;

    v8f acc[4][4];
#pragma unroll
    for (int im = 0; im < 4; ++im)
#pragma unroll
        for (int in_ = 0; in_ < 4; ++in_)
            acc[im][in_] = (v8f){0.f, 0.f, 0.f, 0.f, 0.f, 0.f, 0.f, 0.f};

#if HAVE_TDM
    const unsigned lds_base = (unsigned)(size_t)(void*)&lds[0];
    // prologue: wave 0 issues tile loads for buffer 0 (A then B, in-order per wave)
    if (wv == 0) {
        tdm_tile_load(W + (size_t)bm * N_DIM, lds_base, BM);
        tdm_tile_load(X + (size_t)bn * N_DIM, lds_base + BM * LSTR * 4, BN);
    }
#endif

    for (int it = 0; it < NIT; ++it) {
        const int p = it & 1;
        const float* Ap = &lds[p * BUF];
        const float* Bp = &lds[p * BUF + BM * LSTR];

#if HAVE_TDM
        if (wv == 0) __builtin_amdgcn_s_wait_tensorcnt(0);  // buffer p complete
        __syncthreads();   // p visible to all; 1-p no longer being read by anyone
        if (wv == 0 && it + 1 < NIT) {
            const int k1 = (it + 1) * BK;
            const unsigned nb = lds_base + (unsigned)((1 - p) * BUF) * 4u;
            tdm_tile_load(W + (size_t)bm * N_DIM + k1, nb, BM);
            tdm_tile_load(X + (size_t)bn * N_DIM + k1, nb + BM * LSTR * 4, BN);
        }
#else
        // Fallback: cooperative synchronous tile load into buffer p
        __syncthreads();
        {
            const int k0 = it * BK;
            // (BM+BN) rows x 4 float4 chunks = 1536 float4; 6 per thread
#pragma unroll
            for (int c = 0; c < 6; ++c) {
                const int fidx = tid + 256 * c;
                const int row  = fidx >> 2;
                const int c4   = (fidx & 3) << 2;
                const float* src = (row < BM)
                    ? (W + (size_t)(bm + row) * N_DIM + k0 + c4)
                    : (X + (size_t)(bn + row - BM) * N_DIM + k0 + c4);
                *(float4*)&lds[p * BUF + row * LSTR + c4] = *(const float4*)src;
            }
        }
        __syncthreads();
#endif

#pragma unroll
        for (int kk = 0; kk < BK; kk += 4) {
            v2f af[4], bf[4];
#pragma unroll
            for (int im = 0; im < 4; ++im)
                af[im] = *(const v2f*)&Ap[(wm + im * 16 + r) * LSTR + kk + 2 * hi];
#pragma unroll
            for (int in_ = 0; in_ < 4; ++in_)
                bf[in_] = *(const v2f*)&Bp[(wn + in_ * 16 + r) * LSTR + kk + 2 * hi];
#pragma unroll
            for (int im = 0; im < 4; ++im)
#pragma unroll
                for (int in_ = 0; in_ < 4; ++in_)
                    acc[im][in_] = __builtin_amdgcn_wmma_f32_16x16x4_f32(
                        false, af[im], false, bf[in_],
                        (short)0, acc[im][in_], false, false);
        }
    }

    // Epilogue: C/D 16x16 f32 layout: lanes 0-15 -> M=vgpr, N=lane;
    // lanes 16-31 -> M=8+vgpr, N=lane-16.  bias indexes the BATCH (col) axis.
#pragma unroll
    for (int in_ = 0; in_ < 4; ++in_) {
        const int col = bn + wn + in_ * 16 + r;
        const float bv = bias[col];
#pragma unroll
        for (int im = 0; im < 4; ++im) {
            const int rowbase = bm + wm + im * 16 + hi * 8;
#pragma unroll
            for (int v = 0; v < 8; ++v) {
                Y[(size_t)(rowbase + v) * N_DIM + col] = acc[im][in_][v] + bv;
            }
        }
    }
}

// --------------------------------------------------------------------------
// Host launch. Input order: x, a0..a7, b0..b7, bias.
// d_ws holds the materialized 4096x4096 fp32 w (64 MB).
// --------------------------------------------------------------------------
extern "C" void kernel_launch(void* const* d_in, const int* in_sizes, int n_in,
                              void* d_out, int out_size, void* d_ws, size_t ws_size,
                              hipStream_t stream) {
    (void)in_sizes; (void)n_in; (void)out_size; (void)ws_size;

    const float* x = (const float*)d_in[0];
    const float* a[8];
    const float* b[8];
    for (int i = 0; i < 8; ++i) a[i] = (const float*)d_in[1 + i];
    for (int i = 0; i < 8; ++i) b[i] = (const float*)d_in[9 + i];
    const float* bias = (const float*)d_in[17];

    float* w = (float*)d_ws;   // 4096*4096*4 = 64 MB scratch

    build_w_kernel<<<(N_DIM * (N_DIM / 4)) / 256, 256, 0, stream>>>(
        a[0], a[1], a[2], a[3], a[4], a[5], a[6], a[7],
        b[0], b[1], b[2], b[3], b[4], b[5], b[6], b[7], w);

    dim3 grid(N_DIM / BN, N_DIM / BM);
    kron_gemm_wmma<<<grid, 256, 0, stream>>>(w, x, bias, (float*)d_out);
}